// EvolvingSystem_45329084842224
// MI455X (gfx1250) — compile-verified
//
#include <hip/hip_runtime.h>

typedef float v2f __attribute__((ext_vector_type(2)));
typedef float v8f __attribute__((ext_vector_type(8)));

#define B_SZ   8192
#define K_CL   32
#define D_DIM  128
#define AR_ORD 8
#define T_OUT  24
#define E_DIM  16
#define IN_LEN 96

// ---------------------------------------------------------------------------
// Kernel 1: d2[b,k] = || (mu[k]-z[b]) @ sigma_inv[k] ||^2 via f32 WMMA.
// Block = 128 threads (4 waves). Wave w handles cluster kg+w over a 16-row
// batch tile. 8 e-tiles x 32 chained V_WMMA_F32_16X16X4_F32 per wave.
// ---------------------------------------------------------------------------
__global__ __launch_bounds__(128) void mahal_d2_kernel(
    const float* __restrict__ z,          // [B, D]
    const float* __restrict__ mu,         // [K, D]
    const float* __restrict__ sigma_inv,  // [K, D, D]
    float* __restrict__ d2out)            // [B, K]
{
  __shared__ float zt[16][132];   // padded: bank = (4*row + col) % 64 -> conflict-free column reads
  __shared__ float mus[4][128];

  const int tid = threadIdx.x;
  const int b0  = blockIdx.x * 16;
  const int kg  = blockIdx.y * 4;

  // cooperative stage: z tile (16x128 f32) via float4
  for (int i = tid; i < 16 * 32; i += 128) {
    const int row = i >> 5;
    const int c4  = (i & 31) << 2;
    const float4 v = *(const float4*)(z + (size_t)(b0 + row) * D_DIM + c4);
    zt[row][c4 + 0] = v.x; zt[row][c4 + 1] = v.y;
    zt[row][c4 + 2] = v.z; zt[row][c4 + 3] = v.w;
  }
  // cooperative stage: 4 mu rows
  for (int i = tid; i < 4 * 128; i += 128) {
    mus[i >> 7][i & 127] = mu[(size_t)(kg + (i >> 7)) * D_DIM + (i & 127)];
  }
  __syncthreads();

  const int wave = tid >> 5;
  const int lane = tid & 31;
  const int m    = lane & 15;   // A: row M / B: col N / C: col N
  const int half = lane >> 4;
  const int kkb  = half * 2;    // K-slot base within the 16x16x4 fragment
  const int k    = kg + wave;
  const float* __restrict__ sig = sigma_inv + (size_t)k * D_DIM * D_DIM;

  v8f sq = {};                  // per-element running sum of V^2 across e-tiles
  for (int et = 0; et < 8; ++et) {
    const int e0 = et * 16;
    v8f acc = {};
    #pragma unroll 4
    for (int s = 0; s < 32; ++s) {
      const int d0 = s * 4;
      v2f a, bb;
      // A fragment: A[m, kk] = mu[k, d0+kk] - z[b0+m, d0+kk]
      a.x = mus[wave][d0 + kkb]     - zt[m][d0 + kkb];
      a.y = mus[wave][d0 + kkb + 1] - zt[m][d0 + kkb + 1];
      // B fragment: B[kk, n] = sigma_inv[k, d0+kk, e0+n]   (n == m)
      bb.x = sig[(size_t)(d0 + kkb)     * D_DIM + e0 + m];
      bb.y = sig[(size_t)(d0 + kkb + 1) * D_DIM + e0 + m];
      acc = __builtin_amdgcn_wmma_f32_16x16x4_f32(
          /*neg_a=*/false, a, /*neg_b=*/false, bb,
          /*c_mod=*/(short)0, acc, /*reuse_a=*/false, /*reuse_b=*/false);
    }
    sq += acc * acc;            // elementwise V^2 accumulation
  }

  // C layout: VGPR r -> rows M=r (lanes 0-15) and M=r+8 (lanes 16-31), N=lane%16.
  // Reduce over N inside each 16-lane half; write d2.
  #pragma unroll
  for (int r = 0; r < 8; ++r) {
    float s = sq[r];
    s += __shfl_xor(s, 1, 32);
    s += __shfl_xor(s, 2, 32);
    s += __shfl_xor(s, 4, 32);
    s += __shfl_xor(s, 8, 32);
    if (m == 0) {
      d2out[(size_t)(b0 + half * 8 + r) * K_CL + k] = s;
    }
  }
}

// ---------------------------------------------------------------------------
// Kernel 2: softmax over K, ARIMAX recursion, weighted combine + zero x_recon.
// One wave32 per batch row; lane == cluster index.
// ---------------------------------------------------------------------------
__global__ __launch_bounds__(256) void combine_kernel(
    const float* __restrict__ y,     // [B, IN_LEN]
    const float* __restrict__ u,     // [B, E]
    const float* __restrict__ A,     // [K, ORDER]
    const float* __restrict__ Bx,    // [K, E]
    const float* __restrict__ bias,  // [K]
    const float* __restrict__ d2,    // [B, K]
    float* __restrict__ out)         // [B*T_OUT] y_con, then [B*IN_LEN] x_recon
{
  const int tid   = threadIdx.x;
  const int bbase = blockIdx.x * 8;
  const int b     = bbase + (tid >> 5);
  const int k     = tid & 31;

  // zero-fill x_recon slice for this block's 8 rows
  float* xr = out + (size_t)B_SZ * T_OUT + (size_t)bbase * IN_LEN;
  for (int i = tid; i < 8 * IN_LEN; i += 256) xr[i] = 0.0f;

  // softmax over K of -max(d2, 0)
  float dd = fmaxf(d2[(size_t)b * K_CL + k], 0.0f);
  float nl = -dd;
  float mx = nl;
  #pragma unroll
  for (int off = 16; off >= 1; off >>= 1) mx = fmaxf(mx, __shfl_xor(mx, off, 32));
  float e = __expf(nl - mx);
  float se = e;
  #pragma unroll
  for (int off = 16; off >= 1; off >>= 1) se += __shfl_xor(se, off, 32);
  const float psi = e / se;

  // exogenous drive: ex = u[b] . Bx[k] + bias[k]
  float ex = bias[k];
  #pragma unroll
  for (int j = 0; j < E_DIM; ++j)
    ex += u[(size_t)b * E_DIM + j] * Bx[(size_t)k * E_DIM + j];

  // AR(ORDER) recursion, T_OUT steps
  float hist[AR_ORD], a[AR_ORD];
  #pragma unroll
  for (int o = 0; o < AR_ORD; ++o) {
    hist[o] = y[(size_t)b * IN_LEN + (IN_LEN - AR_ORD) + o];
    a[o]    = A[(size_t)k * AR_ORD + o];
  }

  for (int t = 0; t < T_OUT; ++t) {
    float nxt = ex;
    #pragma unroll
    for (int o = 0; o < AR_ORD; ++o) nxt = fmaf(a[o], hist[o], nxt);
    #pragma unroll
    for (int o = 0; o < AR_ORD - 1; ++o) hist[o] = hist[o + 1];
    hist[AR_ORD - 1] = nxt;

    float c = psi * nxt;
    #pragma unroll
    for (int off = 16; off >= 1; off >>= 1) c += __shfl_xor(c, off, 32);
    if (k == 0) out[(size_t)b * T_OUT + t] = c;
  }
}

// ---------------------------------------------------------------------------
extern "C" void kernel_launch(void* const* d_in, const int* in_sizes, int n_in,
                              void* d_out, int out_size, void* d_ws, size_t ws_size,
                              hipStream_t stream) {
  const float* y         = (const float*)d_in[0];
  const float* z         = (const float*)d_in[1];
  const float* u         = (const float*)d_in[2];
  const float* mu        = (const float*)d_in[3];
  const float* sigma_inv = (const float*)d_in[4];
  const float* A         = (const float*)d_in[5];
  const float* Bx        = (const float*)d_in[6];
  const float* bias      = (const float*)d_in[7];
  float* out = (float*)d_out;
  float* d2  = (float*)d_ws;   // [B, K] f32 = 1 MB scratch

  dim3 g1(B_SZ / 16, K_CL / 4);
  mahal_d2_kernel<<<g1, 128, 0, stream>>>(z, mu, sigma_inv, d2);
  combine_kernel<<<B_SZ / 8, 256, 0, stream>>>(y, u, A, Bx, bias, d2, out);
}